// TtLlamaAttention_67997922230830
// MI455X (gfx1250) — compile-verified
//
#include <hip/hip_runtime.h>
#include <math.h>
#include <stdint.h>

typedef float v2f __attribute__((ext_vector_type(2)));
typedef float v4f __attribute__((ext_vector_type(4)));
typedef float v8f __attribute__((ext_vector_type(8)));
typedef unsigned int v4u __attribute__((ext_vector_type(4)));
typedef int v4i __attribute__((ext_vector_type(4)));
typedef int v8i __attribute__((ext_vector_type(8)));
typedef int i32x4 __attribute__((vector_size(16)));   // matches builtin param type

#define DIMSZ   8192
#define NHEAD   64
#define NKVH    8
#define HDIM    128
#define BATCH   32
#define MAXSEQ  4096
#define SPOS    2047
#define LKEYS   2048
#define KSPLIT  8

// CDNA5 data-mover availability (device pass only; host pass sees 0, so any
// host-visible sizing below must NOT depend on these).
#if __has_builtin(__builtin_amdgcn_global_load_async_to_lds_b128) && \
    __has_builtin(__builtin_amdgcn_s_wait_asynccnt)
#define HAS_ASYNC 1
#else
#define HAS_ASYNC 0
#endif
#if __has_builtin(__builtin_amdgcn_tensor_load_to_lds) && \
    __has_builtin(__builtin_amdgcn_s_wait_tensorcnt)
#define HAS_TDM 1
#else
#define HAS_TDM 0
#endif

// D = A(16x4) * B(4x16) + C, fp32: keeps reference numerics; the layer is
// HBM-bound (~8 FLOP/B) so f32 WMMA rate is not the limiter.
static __device__ __forceinline__ v8f wmma_f32(v2f a, v2f b, v8f c) {
  return __builtin_amdgcn_wmma_f32_16x16x4_f32(false, a, false, b, (short)0, c,
                                               false, false);
}

#if HAS_ASYNC
// 16B per lane, global -> LDS through the async path (ASYNCcnt tracked).
// Builtin params are (v4i AS(1)*, v4i AS(3)*, imm offset, imm cpol); the LDS
// generic pointer truncates to its 32-bit LDS byte offset.
static __device__ __forceinline__ void async_cp_b128(const float* g, float* l) {
  __builtin_amdgcn_global_load_async_to_lds_b128(
      (__attribute__((address_space(1))) i32x4*)(uintptr_t)g,
      (__attribute__((address_space(3))) i32x4*)(unsigned int)(uintptr_t)l,
      0, 0);
}
#endif

#if HAS_TDM
// 1-D TDM copy: 2048 contiguous f32 (one 16-row x 128-col K-cache tile).
// D# group0: count=1, lds_addr, 57-bit global addr, type=2 ("image").
// D# group1: data_size=4B, tensor_dim0=2048, tensor_dim1=1, tile_dim0=2048,
//            tile_dim1=1, tensor_dim0_stride=2048 (fields packed per ISA 8.4).
// This toolchain exposes the 6-arg builtin form: (g0, g1, g2, g3, g4, cpol).
static __device__ __forceinline__ void tdm_load_row2048(const float* gsrc,
                                                        float* ldst) {
  unsigned long long ga = (unsigned long long)(uintptr_t)gsrc;
  unsigned int la = (unsigned int)(uintptr_t)ldst;
  v4u g0 = { 1u, la, (unsigned int)ga,
             ((unsigned int)((ga >> 32) & 0x01FFFFFFull)) | (2u << 30) };
  v8i g1 = { 0x20000,            // data_size = 2 (4 bytes)
             0x08000000,         // tensor_dim0[15:0]=2048 in bits [31:16]
             0x00010000,         // tensor_dim0[31:16]=0, tensor_dim1[15:0]=1
             0x08000000,         // tensor_dim1 hi=0, tile_dim0=2048
             1,                  // tile_dim1=1, tile_dim2=0
             2048,               // tensor_dim0_stride lo
             0, 0 };
  v4i z4 = { 0, 0, 0, 0 };
  v8i z8 = { 0, 0, 0, 0, 0, 0, 0, 0 };
  __builtin_amdgcn_tensor_load_to_lds(g0, g1, z4, z4, z8, 0);
}
#endif

__global__ void zero_f32(float* __restrict__ p, int n) {
  int i = blockIdx.x * blockDim.x + threadIdx.x;
  if (i < n) p[i] = 0.0f;
}

// Y(32xN) += X(32xK) * W(KxN) over the K-range owned by blockIdx.y.
// One wave per block; wave owns a 32x16 output tile (two 16x16 WMMA accums).
// 32-deep K chunks are double-buffered in LDS via async global->LDS copies,
// software-pipelined on ASYNCcnt; padded LDS strides keep fragment reads
// bank-conflict free.
__global__ __launch_bounds__(32)
void gemm32_splitk(const float* __restrict__ X, int K,
                   const float* __restrict__ W, int N,
                   float* __restrict__ Y, int kchunk)
{
  const int n0   = blockIdx.x * 16;
  const int kbeg = blockIdx.y * kchunk;
  const int lane = threadIdx.x;
  const int half = lane >> 4;   // 0: K={0,1}, 1: K={2,3} of each 4-group
  const int l16  = lane & 15;

  __shared__ float sX[2][32 * 36];  // 32 rows x 32 k, stride 36 (16B-aligned)
  __shared__ float sW[2][32 * 20];  // 32 k   x 16 n, stride 20 (16B-aligned)

  const int xrow = lane >> 3;        // + i*4
  const int xc4  = (lane & 7) * 4;
  const int wrow = lane >> 2;        // + i*8
  const int wc4  = (lane & 3) * 4;

  v8f acc0 = {0.f,0.f,0.f,0.f,0.f,0.f,0.f,0.f};
  v8f acc1 = {0.f,0.f,0.f,0.f,0.f,0.f,0.f,0.f};

  auto stage = [&](int buf, int kc) {
#if HAS_ASYNC
    #pragma unroll
    for (int i = 0; i < 8; ++i)
      async_cp_b128(X + (size_t)(i * 4 + xrow) * K + kc + xc4,
                    &sX[buf][(i * 4 + xrow) * 36 + xc4]);
    #pragma unroll
    for (int i = 0; i < 4; ++i)
      async_cp_b128(W + (size_t)(kc + i * 8 + wrow) * N + n0 + wc4,
                    &sW[buf][(i * 8 + wrow) * 20 + wc4]);
#else
    #pragma unroll
    for (int i = 0; i < 8; ++i) {
      v4f v = *(const v4f*)(X + (size_t)(i * 4 + xrow) * K + kc + xc4);
      *(v4f*)(&sX[buf][(i * 4 + xrow) * 36 + xc4]) = v;
    }
    #pragma unroll
    for (int i = 0; i < 4; ++i) {
      v4f v = *(const v4f*)(W + (size_t)(kc + i * 8 + wrow) * N + n0 + wc4);
      *(v4f*)(&sW[buf][(i * 8 + wrow) * 20 + wc4]) = v;
    }
#endif
  };

  const int nch = kchunk / 32;
  stage(0, kbeg);
  for (int c = 0; c < nch; ++c) {
    const int cur = c & 1;
    if (c + 1 < nch) {
      stage(cur ^ 1, kbeg + (c + 1) * 32);   // prefetch next chunk
#if HAS_ASYNC
      __builtin_amdgcn_s_wait_asynccnt(12);  // 12 in flight = current chunk done
#endif
    } else {
#if HAS_ASYNC
      __builtin_amdgcn_s_wait_asynccnt(0);
#endif
    }
    asm volatile("" ::: "memory");
    __syncthreads();

    #pragma unroll
    for (int s = 0; s < 8; ++s) {
      int kk = s * 4 + half * 2;
      v2f a0 = { sX[cur][l16 * 36 + kk],        sX[cur][l16 * 36 + kk + 1] };
      v2f a1 = { sX[cur][(16 + l16) * 36 + kk], sX[cur][(16 + l16) * 36 + kk + 1] };
      v2f b  = { sW[cur][kk * 20 + l16],        sW[cur][(kk + 1) * 20 + l16] };
      acc0 = wmma_f32(a0, b, acc0);
      acc1 = wmma_f32(a1, b, acc1);
    }
  }

  // Split-K epilogue: fp32 global atomics into pre-zeroed accumulator.
  #pragma unroll
  for (int v = 0; v < 8; ++v) {
    int row = v + half * 8;                       // D layout: rows v / v+8
    atomicAdd(&Y[(size_t)row * N + n0 + l16],        acc0[v]);
    atomicAdd(&Y[(size_t)(16 + row) * N + n0 + l16], acc1[v]);
  }
}

// One workgroup per (batch, kv-head): 8 grouped Q heads vs 2048 keys.
// RoPE folded into Q/K-new load; 1/sqrt(HD) folded into Q; mask is provably
// zero (START_POS+1 == L). The cache is never written: the l==SPOS row is
// substituted from the freshly projected k/v at load time.
// K tiles are streamed by the Tensor Data Mover (double-buffered per wave,
// pipelined on TENSORcnt) when available.
__global__ __launch_bounds__(128)
void attn_kernel(const float* __restrict__ q_raw,   // 32 x 64 x 128
                 const float* __restrict__ k_raw,   // 32 x 8 x 128
                 const float* __restrict__ v_raw,   // 32 x 8 x 128
                 const float* __restrict__ rot,     // 1 x 32 x 128 x 128
                 const float* __restrict__ cache_k, // 32 x 8 x 4096 x 128
                 const float* __restrict__ cache_v,
                 float* __restrict__ attn_out)      // 32 x 64 x 128
{
  extern __shared__ float smem[];
  float* sS  = smem;                  // 8 x 2048 scores/probs (64 KB)
  float* sQ  = sS + 8 * LKEYS;        // 16 x 128 rotated+scaled Q (rows 8..15 = 0)
  float* sK  = sQ + 16 * HDIM;        // 128 rotated new k row
  float* sKt = sK + HDIM;             // 4 waves x 2 bufs x (16 x 128) K tiles

  const int b    = blockIdx.x >> 3;
  const int kv   = blockIdx.x & 7;
  const int tid  = threadIdx.x;
  const int lane = tid & 31;
  const int wave = tid >> 5;
  const int half = lane >> 4;
  const int l16  = lane & 15;

  const float* Rb = rot + (size_t)b * HDIM * HDIM;
  const float qscale = 0.08838834764831845f;  // 1/sqrt(128)

  // RoPE(Q): out[2i] = x[2i]*c - x[2i+1]*s ; out[2i+1] = x[2i]*s + x[2i+1]*c,
  // with c = R[2i][2i], s = R[2i][2i+1] read straight from rot_mat.
  for (int i = tid; i < 8 * 64; i += 128) {
    int g = i >> 6, e = (i & 63) * 2;
    const float* qr = q_raw + (size_t)((b * NHEAD) + kv * 8 + g) * HDIM;
    float c = Rb[(size_t)e * HDIM + e], s = Rb[(size_t)e * HDIM + e + 1];
    float x0 = qr[e], x1 = qr[e + 1];
    sQ[g * HDIM + e]     = (x0 * c - x1 * s) * qscale;
    sQ[g * HDIM + e + 1] = (x0 * s + x1 * c) * qscale;
  }
  for (int i = tid; i < 8 * HDIM; i += 128) sQ[8 * HDIM + i] = 0.0f;  // pad rows
  if (tid < 64) {
    int e = tid * 2;
    const float* kr = k_raw + (size_t)(b * NKVH + kv) * HDIM;
    float c = Rb[(size_t)e * HDIM + e], s = Rb[(size_t)e * HDIM + e + 1];
    float x0 = kr[e], x1 = kr[e + 1];
    sK[e]     = x0 * c - x1 * s;
    sK[e + 1] = x0 * s + x1 * c;
  }
  __syncthreads();

  const float* Kb = cache_k + (size_t)(b * NKVH + kv) * MAXSEQ * HDIM;
  const float* Vb = cache_v + (size_t)(b * NKVH + kv) * MAXSEQ * HDIM;

  // Pass 1: S = Q * K^T, 16-key tiles round-robined over the 4 waves.
#if HAS_TDM
  {
    float* myT = sKt + wave * (2 * 16 * HDIM);
    tdm_load_row2048(Kb + (size_t)(wave * 16) * HDIM, myT);  // prime buf 0
    int c = 0;
    for (int j = wave; j < LKEYS / 16; j += 4, ++c) {
      const int cur = c & 1;
      if (j + 4 < LKEYS / 16) {
        tdm_load_row2048(Kb + (size_t)((j + 4) * 16) * HDIM,
                         myT + (cur ^ 1) * 16 * HDIM);       // prefetch next
        __builtin_amdgcn_s_wait_tensorcnt(1);
      } else {
        __builtin_amdgcn_s_wait_tensorcnt(0);
      }
      asm volatile("" ::: "memory");
      const float* Kt = myT + cur * 16 * HDIM;
      int l0  = j * 16;
      int myl = l0 + l16;
      v8f acc = {0.f,0.f,0.f,0.f,0.f,0.f,0.f,0.f};
      #pragma unroll 4
      for (int k0 = 0; k0 < HDIM; k0 += 4) {
        int kk = k0 + half * 2;
        v2f a  = { sQ[l16 * HDIM + kk], sQ[l16 * HDIM + kk + 1] };
        v2f bf = { Kt[l16 * HDIM + kk], Kt[l16 * HDIM + kk + 1] };
        if (myl == SPOS) { bf.x = sK[kk]; bf.y = sK[kk + 1]; }  // new token row
        acc = wmma_f32(a, bf, acc);
      }
      if (half == 0) {
        #pragma unroll
        for (int v = 0; v < 8; ++v) sS[v * LKEYS + l0 + l16] = acc[v];
      }
    }
  }
#else
  for (int j = wave; j < LKEYS / 16; j += 4) {
    int l0  = j * 16;
    int myl = l0 + l16;
    const float* Krow = Kb + (size_t)myl * HDIM;
    v8f acc = {0.f,0.f,0.f,0.f,0.f,0.f,0.f,0.f};
    #pragma unroll 4
    for (int k0 = 0; k0 < HDIM; k0 += 4) {
      int kk = k0 + half * 2;
      v2f a  = { sQ[l16 * HDIM + kk], sQ[l16 * HDIM + kk + 1] };
      v2f bf = { Krow[kk], Krow[kk + 1] };
      if (myl == SPOS) { bf.x = sK[kk]; bf.y = sK[kk + 1]; }
      acc = wmma_f32(a, bf, acc);
    }
    if (half == 0) {
      #pragma unroll
      for (int v = 0; v < 8; ++v) sS[v * LKEYS + l0 + l16] = acc[v];
    }
  }
#endif
  __syncthreads();

  // Pass 2: softmax per row; 16 threads per row, shfl_xor reductions.
  {
    int row = tid >> 4, t = tid & 15;
    float* Sr = sS + row * LKEYS;
    float m = -3.402823466e38f;
    for (int i = t; i < LKEYS; i += 16) m = fmaxf(m, Sr[i]);
    #pragma unroll
    for (int off = 8; off > 0; off >>= 1) m = fmaxf(m, __shfl_xor(m, off, 16));
    float sum = 0.0f;
    for (int i = t; i < LKEYS; i += 16) { float e = __expf(Sr[i] - m); Sr[i] = e; sum += e; }
    #pragma unroll
    for (int off = 8; off > 0; off >>= 1) sum += __shfl_xor(sum, off, 16);
    float inv = 1.0f / sum;
    for (int i = t; i < LKEYS; i += 16) Sr[i] *= inv;
  }
  __syncthreads();

  // Pass 3: O = P * V, 16-col output tiles over the 4 waves.
  const float* vnew = v_raw + (size_t)(b * NKVH + kv) * HDIM;
  for (int tt = wave; tt < HDIM / 16; tt += 4) {
    int d0 = tt * 16;
    v8f acc = {0.f,0.f,0.f,0.f,0.f,0.f,0.f,0.f};
    #pragma unroll 4
    for (int l0 = 0; l0 < LKEYS; l0 += 4) {
      int lb = l0 + half * 2;
      int am = l16 & 7;                         // rows 8..15 of D are discarded
      v2f a = { sS[am * LKEYS + lb], sS[am * LKEYS + lb + 1] };
      const float* V0 = Vb + (size_t)lb * HDIM;
      v2f bf = { V0[d0 + l16], V0[HDIM + d0 + l16] };
      if (lb == SPOS)     bf.x = vnew[d0 + l16];
      if (lb + 1 == SPOS) bf.y = vnew[d0 + l16];
      acc = wmma_f32(a, bf, acc);
    }
    if (half == 0) {
      #pragma unroll
      for (int v = 0; v < 8; ++v)
        attn_out[(size_t)(b * NHEAD + kv * 8 + v) * HDIM + d0 + l16] = acc[v];
    }
  }
}

extern "C" void kernel_launch(void* const* d_in, const int* in_sizes, int n_in,
                              void* d_out, int out_size, void* d_ws, size_t ws_size,
                              hipStream_t stream) {
  const float* x       = (const float*)d_in[0];   // 1x1x32x8192
  const float* rot     = (const float*)d_in[1];   // 1x32x128x128
  // d_in[2] = attn_mask: identically zero for this config (L == START_POS+1)
  const float* wq      = (const float*)d_in[3];   // 8192x8192
  const float* wk      = (const float*)d_in[4];   // 8192x1024
  const float* wv      = (const float*)d_in[5];   // 8192x1024
  const float* wo      = (const float*)d_in[6];   // 8192x8192
  const float* cache_k = (const float*)d_in[7];   // 32x8x4096x128
  const float* cache_v = (const float*)d_in[8];   // 32x8x4096x128
  float* out = (float*)d_out;                     // 1x1x32x8192

  // Workspace layout (floats)
  float* ws       = (float*)d_ws;
  float* raw_q    = ws;                         // 32*64*128 = 262144
  float* raw_k    = raw_q + 262144;             // 32*8*128  = 32768
  float* raw_v    = raw_k + 32768;              // 32768
  float* attn_out = raw_v + 32768;              // 262144
  const int n_qkv = 262144 + 32768 + 32768;

  // Zero split-K accumulators (ws is poisoned) and d_out.
  zero_f32<<<(n_qkv + 255) / 256, 256, 0, stream>>>(raw_q, n_qkv);
  zero_f32<<<(262144 + 255) / 256, 256, 0, stream>>>(out, 262144);

  // QKV projections: X(32x8192) * W, 8-way split-K.
  const int kchunk = DIMSZ / KSPLIT;
  gemm32_splitk<<<dim3(DIMSZ / 16, KSPLIT), 32, 0, stream>>>(x, DIMSZ, wq, DIMSZ, raw_q, kchunk);
  gemm32_splitk<<<dim3(NKVH * HDIM / 16, KSPLIT), 32, 0, stream>>>(x, DIMSZ, wk, NKVH * HDIM, raw_k, kchunk);
  gemm32_splitk<<<dim3(NKVH * HDIM / 16, KSPLIT), 32, 0, stream>>>(x, DIMSZ, wv, NKVH * HDIM, raw_v, kchunk);

  // Attention: one WG per (b, kv). Dynamic LDS sized for the max (TDM) layout
  // unconditionally: 64KB scores + 8.5KB Q/Knew + 64KB K-tile double buffers.
  size_t smem = (size_t)(8 * LKEYS + 16 * HDIM + HDIM + 4 * 2 * 16 * HDIM) * sizeof(float);
  (void)hipFuncSetAttribute((const void*)attn_kernel,
                            hipFuncAttributeMaxDynamicSharedMemorySize, (int)smem);
  attn_kernel<<<BATCH * NKVH, 128, smem, stream>>>(raw_q, raw_k, raw_v, rot,
                                                   cache_k, cache_v, attn_out);

  // Output projection: attn_out(32x8192) * wo(8192x8192) -> d_out.
  gemm32_splitk<<<dim3(DIMSZ / 16, KSPLIT), 32, 0, stream>>>(attn_out, DIMSZ, wo, DIMSZ, out, kchunk);
}